// RPNTrainingProposal_21784074125318
// MI455X (gfx1250) — compile-verified
//
#include <hip/hip_runtime.h>
#include <hip/hip_bf16.h>
#include <cstdint>

#define NG 100            // number of GT boxes
#define POS_T 0.7f
#define NEG_T 0.3f
#define TOTAL_S 256
#define MAX_POS 128

typedef unsigned int u32x4 __attribute__((ext_vector_type(4)));
typedef int          i32x8 __attribute__((ext_vector_type(8)));
typedef int          i32x4 __attribute__((ext_vector_type(4)));

__device__ __forceinline__ unsigned ballot32(bool p) {
#if __has_builtin(__builtin_amdgcn_ballot_w32)
    return __builtin_amdgcn_ballot_w32(p);
#else
    return (unsigned)__ballot(p);
#endif
}

// ---------------------------------------------------------------------------
// Kernel 0: zero the pos/neg counters (deterministic per launch, no stale ws)
// ---------------------------------------------------------------------------
__global__ void rpn_zero_counters(int* counters) {
    counters[0] = 0;
    counters[1] = 0;
}

// ---------------------------------------------------------------------------
// Kernel 1: per-anchor max IoU vs 100 GT boxes, labels, pos/neg census.
// GT table (1600 B) is DMA'd into LDS offset 0 via the Tensor Data Mover.
// Two anchors per thread: each LDS broadcast read feeds two independent
// min/max dependency chains (better VALU IPC, half the DS issue per pair).
// ---------------------------------------------------------------------------
__global__ void __launch_bounds__(256)
rpn_iou_label_kernel(const float4* __restrict__ anchors,
                     const float*  __restrict__ gt,       // NG*4 floats
                     float*        __restrict__ scores,
                     int*          __restrict__ labels,
                     int*          __restrict__ counters,
                     int n) {
    // single shared allocation -> its LDS byte offset is 0 (TDM lds_addr = 0)
    __shared__ float smem[NG * 4 + NG];   // [0..399] gt boxes, [400..499] areas
    float* s_gt   = smem;
    float* s_area = smem + NG * 4;

#if defined(__gfx1250__) && __has_builtin(__builtin_amdgcn_tensor_load_to_lds)
    if (threadIdx.x < 32) {               // one wave issues the TDM descriptor
        unsigned long long ga = (unsigned long long)(uintptr_t)gt;
        u32x4 g0;
        g0.x = 1u;                                            // count=1, user D#
        g0.y = 0u;                                            // lds_addr = 0
        g0.z = (unsigned)(ga & 0xFFFFFFFFull);                // global_addr[31:0]
        g0.w = (unsigned)((ga >> 32) & 0x01FFFFFFull)         // global_addr[56:32]
               | (2u << 30);                                  // type = 2 (image)
        i32x8 g1;
        g1[0] = 0x00020000;          // workgroup_mask=0, data_size=2 (4 bytes)
        g1[1] = (int)(400u << 16);   // tensor_dim0[15:0] = 400 (elements)
        g1[2] = (int)(1u << 16);     // tensor_dim0 hi=0, tensor_dim1[15:0]=1
        g1[3] = (int)(400u << 16);   // tensor_dim1 hi=0, tile_dim0=400
        g1[4] = 1;                   // tile_dim1=1, tile_dim2=0 (unused)
        g1[5] = 400;                 // tensor_dim0_stride[31:0] = 400
        g1[6] = (int)(400u << 16);   // stride0 hi=0, tensor_dim1_stride[15:0]=400
        g1[7] = 0;                   // tensor_dim1_stride hi = 0
        i32x4 g2; g2[0] = 1; g2[1] = 1; g2[2] = 0; g2[3] = 0; // dims2/3=1, unused
        i32x4 g3; g3[0] = 0; g3[1] = 0; g3[2] = 0; g3[3] = 0; // tile_dim3/4=0
        i32x8 g4; g4[0] = 0; g4[1] = 0; g4[2] = 0; g4[3] = 0; // extra group (6-arg
        g4[4] = 0; g4[5] = 0; g4[6] = 0; g4[7] = 0;           // clang-23 form)
        __builtin_amdgcn_tensor_load_to_lds(g0, g1, g2, g3, g4, 0);
        __builtin_amdgcn_s_wait_tensorcnt(0);
    }
    __syncthreads();
#else
    for (int t = threadIdx.x; t < NG * 4; t += blockDim.x) s_gt[t] = gt[t];
    __syncthreads();
#endif

    // precompute GT areas once per block
    for (int j = threadIdx.x; j < NG; j += blockDim.x) {
        float x1 = s_gt[4 * j + 0], y1 = s_gt[4 * j + 1];
        float x2 = s_gt[4 * j + 2], y2 = s_gt[4 * j + 3];
        s_area[j] = (x2 - x1) * (y2 - y1);
    }
    __syncthreads();

    const int tcount = gridDim.x * blockDim.x;
    for (int i0 = blockIdx.x * blockDim.x + threadIdx.x; i0 < n; i0 += 2 * tcount) {
        const int  i1   = i0 + tcount;
        const bool has1 = (i1 < n);
        float4 a0 = anchors[i0];                     // global_load_b128, coalesced
        float4 a1 = has1 ? anchors[i1] : a0;
        float areaA0 = (a0.z - a0.x) * (a0.w - a0.y);
        float areaA1 = (a1.z - a1.x) * (a1.w - a1.y);
        // track max(inter/den) without per-pair division: cross-multiply compare
        float bN0 = 0.0f, bD0 = 1.0f;
        float bN1 = 0.0f, bD1 = 1.0f;
#pragma unroll 2
        for (int j = 0; j < NG; ++j) {
            const float bx1 = s_gt[4 * j + 0], by1 = s_gt[4 * j + 1];
            const float bx2 = s_gt[4 * j + 2], by2 = s_gt[4 * j + 3];
            const float ab  = s_area[j];
            // anchor 0
            float iw0 = fmaxf(fminf(a0.z, bx2) - fmaxf(a0.x, bx1), 0.0f);
            float ih0 = fmaxf(fminf(a0.w, by2) - fmaxf(a0.y, by1), 0.0f);
            float it0 = iw0 * ih0;
            float dn0 = areaA0 + ab - it0;           // > 0 always
            bool b0 = it0 * bD0 > bN0 * dn0;
            bN0 = b0 ? it0 : bN0;
            bD0 = b0 ? dn0 : bD0;
            // anchor 1 (independent chain)
            float iw1 = fmaxf(fminf(a1.z, bx2) - fmaxf(a1.x, bx1), 0.0f);
            float ih1 = fmaxf(fminf(a1.w, by2) - fmaxf(a1.y, by1), 0.0f);
            float it1 = iw1 * ih1;
            float dn1 = areaA1 + ab - it1;
            bool b1 = it1 * bD1 > bN1 * dn1;
            bN1 = b1 ? it1 : bN1;
            bD1 = b1 ? dn1 : bD1;
        }
        float sc0 = bN0 / bD0;                       // one divide per anchor
        float sc1 = bN1 / bD1;
        int lab0 = (sc0 > POS_T) ? 1 : ((sc0 < NEG_T) ? 0 : -1);
        int lab1 = (sc1 > POS_T) ? 1 : ((sc1 < NEG_T) ? 0 : -1);
        scores[i0] = sc0;
        labels[i0] = lab0;
        if (has1) { scores[i1] = sc1; labels[i1] = lab1; }
        if (!has1) lab1 = -2;                        // exclude from census

        // wave32 aggregated census: 2 atomics per wave per pair of anchors
        unsigned bp = ballot32(lab0 == 1);
        unsigned bn = ballot32(lab0 == 0);
        unsigned bp2 = ballot32(lab1 == 1);
        unsigned bn2 = ballot32(lab1 == 0);
        if ((threadIdx.x & 31) == 0) {
            int cp = __popc(bp) + __popc(bp2);
            int cn = __popc(bn) + __popc(bn2);
            if (cp) atomicAdd(&counters[0], cp);
            if (cn) atomicAdd(&counters[1], cn);
        }
    }
}

// ---------------------------------------------------------------------------
// Kernel 2: single-block deterministic sampling/compaction.
// out layout (float): [0..255] sel idx, [256..511] labels, [512..767] scores
// ---------------------------------------------------------------------------
__global__ void __launch_bounds__(1024)
rpn_sample_kernel(const float* __restrict__ scores,
                  const int*   __restrict__ labels,
                  const int*   __restrict__ counters,
                  float*       __restrict__ out,
                  int n) {
    __shared__ int s_pos[32];
    __shared__ int s_neg[32];
    __shared__ int s_run[2];

    const int tid  = threadIdx.x;
    const int lane = tid & 31;
    const int wid  = tid >> 5;

    const int total_pos = counters[0];
    const int total_neg = counters[1];
    const int cur_pos = min(total_pos, MAX_POS);
    const int cur_neg = min(TOTAL_S - cur_pos, total_neg);

    if (tid < TOTAL_S) {                 // defaults for unused slots
        out[tid]               = -1.0f;  // sel = -1
        out[TOTAL_S + tid]     = -1.0f;  // label = -1
        out[2 * TOTAL_S + tid] = 0.0f;   // score = 0
    }
    if (tid == 0) { s_run[0] = 0; s_run[1] = 0; }
    __syncthreads();

    const unsigned ltmask = (1u << lane) - 1u;
    for (int base = 0; base < n; base += (int)blockDim.x) {
        const int i  = base + tid;
        const int lab = (i < n) ? labels[i] : -2;
        const bool isp = (lab == 1);
        const bool isn = (lab == 0);
        const unsigned bp = ballot32(isp);
        const unsigned bn = ballot32(isn);
        if (lane == 0) { s_pos[wid] = __popc(bp); s_neg[wid] = __popc(bn); }
        __syncthreads();
        if (tid == 0) {                  // exclusive scan of 32 wave totals
            int accp = s_run[0], accn = s_run[1];
            for (int w = 0; w < 32; ++w) {
                int tp = s_pos[w]; s_pos[w] = accp; accp += tp;
                int tn = s_neg[w]; s_neg[w] = accn; accn += tn;
            }
            s_run[0] = accp; s_run[1] = accn;
        }
        __syncthreads();
        if (isp) {
            int r = s_pos[wid] + __popc(bp & ltmask);
            if (r < cur_pos) {
                out[r]               = (float)i;
                out[TOTAL_S + r]     = 1.0f;
                out[2 * TOTAL_S + r] = scores[i];
            }
        }
        if (isn) {
            int r = s_neg[wid] + __popc(bn & ltmask);
            if (r < cur_neg) {
                int s = cur_pos + r;
                out[s]               = (float)i;
                out[TOTAL_S + s]     = 0.0f;
                out[2 * TOTAL_S + s] = scores[i];
            }
        }
        __syncthreads();
        if (s_run[0] >= cur_pos && s_run[1] >= cur_neg) break; // uniform
    }
}

// ---------------------------------------------------------------------------
extern "C" void kernel_launch(void* const* d_in, const int* in_sizes, int n_in,
                              void* d_out, int out_size, void* d_ws, size_t ws_size,
                              hipStream_t stream) {
    const float4* anchors = (const float4*)d_in[0];         // [N,4] f32
    const float*  gt      = (const float*)d_in[1];          // [100,4] f32
    const int n = in_sizes[0] / 4;

    // workspace: scores f32[N] | labels i32[N] | counters i32[2]
    float* scores   = (float*)d_ws;
    int*   labels   = (int*)((char*)d_ws + (size_t)n * sizeof(float));
    int*   counters = (int*)((char*)d_ws + (size_t)n * 2 * sizeof(float));

    rpn_zero_counters<<<1, 1, 0, stream>>>(counters);

    const int threads = 256;
    int blocks = (n + threads * 8 - 1) / (threads * 8);     // ~8 anchors/thread
    if (blocks < 1) blocks = 1;
    if (blocks > 2048) blocks = 2048;
    rpn_iou_label_kernel<<<blocks, threads, 0, stream>>>(anchors, gt, scores,
                                                         labels, counters, n);

    rpn_sample_kernel<<<1, 1024, 0, stream>>>(scores, labels, counters,
                                              (float*)d_out, n);
}